// DecoupledGCNUnit_24713241822099
// MI455X (gfx1250) — compile-verified
//
#include <hip/hip_runtime.h>

typedef __attribute__((ext_vector_type(2))) float v2f;
typedef __attribute__((ext_vector_type(8))) float v8f;

#define N_      64
#define CIN     64
#define T_      256
#define V_      25
#define S_      3
#define COUT    128
#define CING    8      // CIN / GROUPS
#define OTOT    384    // COUT * S
#define COUTG   48     // OTOT / GROUPS
#define KTOT    75     // S * V
#define KP      76     // padded K stride (even -> 8B aligned pairs)
#define TT      2      // timesteps per block
#define TV      (TT * V_)   // 50
#define EPS_    1e-5f

static __device__ __forceinline__ v2f lds2(const float* p) {
    return *reinterpret_cast<const v2f*>(p);
}

// ---------------------------------------------------------------------------
// Kernel 1: fused grouped 1x1 conv + per-subset adjacency contraction.
//   Phase 1: conv -> U[t][c][KP] in LDS (k = s*25+v, k-contiguous)
//   Phase 2: per (n,t): (128 x 75) @ (75 x 25) via v_wmma_f32_16x16x4_f32,
//            A and B fragments are single ds_load_b64 each.
// ---------------------------------------------------------------------------
__global__ __launch_bounds__(256) void k1_fused(
    const float* __restrict__ x,  const float* __restrict__ A,
    const float* __restrict__ PA, const float* __restrict__ Wg,
    const float* __restrict__ bg, float* __restrict__ out)
{
    __shared__ float xs[CIN * TV];       //  3200 f : x tile [cin][t][v]
    __shared__ float wl[OTOT * CING];    //  3072 f : conv weights
    __shared__ float bl[OTOT];           //   384 f : conv bias
    __shared__ float aeT[32 * KP];       //  2432 f : Aeff^T padded [w][k]
    __shared__ float Ul[TT * COUT * KP]; // 19456 f : conv output [t][c][k]

    const int tid  = threadIdx.x;
    const int n    = blockIdx.x >> 7;          // / (T_/TT)
    const int t0   = (blockIdx.x & 127) * TT;

    // ---- stage inputs ----
    for (int u = tid; u < CIN * TV; u += 256) {
        int cin = u / TV;
        int r   = u - cin * TV;
        xs[u] = x[(n * CIN + cin) * (T_ * V_) + t0 * V_ + r];
    }
    for (int u = tid; u < OTOT * CING; u += 256) wl[u] = Wg[u];
    for (int u = tid; u < OTOT;        u += 256) bl[u] = bg[u];
    for (int u = tid; u < 32 * KP; u += 256) {
        int w = u / KP;
        int k = u - w * KP;
        float val = 0.f;
        if (k < KTOT && w < V_) val = A[k * V_ + w] + PA[k * V_ + w];
        aeT[u] = val;
    }
    __syncthreads();

    // ---- Phase 1: grouped conv. One thread per (t,c) row of U. ----
    {
        const int tl = tid >> 7;           // 0..1
        const int c  = tid & 127;          // channel
        float* urow = &Ul[tid * KP];       // (tl*128+c)*KP == tid*KP
        #pragma unroll
        for (int s = 0; s < S_; ++s) {
            const int o = s * COUT + c;
            const int g = o / COUTG;
            const float* wr = &wl[o * CING];
            float wreg[CING];
            #pragma unroll
            for (int i = 0; i < CING; ++i) wreg[i] = wr[i];
            const float bias = bl[o];
            const float* xv = &xs[(g * CING) * TV + tl * V_];
            for (int v = 0; v < V_; ++v) {
                float u = bias;
                #pragma unroll
                for (int i = 0; i < CING; ++i) u += xv[i * TV + v] * wreg[i];
                urow[s * V_ + v] = u;
            }
        }
        urow[KTOT] = 0.f;                  // k = 75 zero pad
    }
    __syncthreads();

    // ---- Phase 2: WMMA. 8 waves x 4 D-tiles = 2t x 2wt x 8ct. ----
    const int lane  = tid & 31;
    const int warp  = tid >> 5;
    const int khalf = lane >> 4;           // lanes 16-31 hold K+2,K+3
    const int col   = lane & 15;
    const int tsel  = warp & 1;
    const int wt    = (warp >> 1) & 1;
    const int ctg   = warp >> 2;           // 0 or 1 -> ct groups of 4
    const int t     = t0 + tsel;
    const int wcol  = wt * 16 + col;
    const float* brow = &aeT[wcol * KP];   // B column for this lane

    for (int cti = 0; cti < 4; ++cti) {
        const int ct = ctg * 4 + cti;
        const int c  = ct * 16 + col;
        const float* arow = &Ul[(tsel * COUT + c) * KP];
        v8f acc = {0.f, 0.f, 0.f, 0.f, 0.f, 0.f, 0.f, 0.f};
        #pragma unroll
        for (int j = 0; j < 19; ++j) {     // K = 75 (+1 pad) in steps of 4
            const int kb = 4 * j + 2 * khalf;
            const v2f a = lds2(arow + kb); // ds_load_b64
            const v2f b = lds2(brow + kb); // ds_load_b64
            acc = __builtin_amdgcn_wmma_f32_16x16x4_f32(
                false, a, false, b, (short)0, acc, false, false);
        }
        // D layout: VGPR r -> row r (lanes 0-15) / row r+8 (lanes 16-31)
        if (wcol < V_) {
            #pragma unroll
            for (int r = 0; r < 8; ++r) {
                const int cc = ct * 16 + r + 8 * khalf;
                out[((n * COUT + cc) * T_ + t) * V_ + wcol] = acc[r];
            }
        }
    }
}

// ---------------------------------------------------------------------------
// Kernel 2: per-channel batch stats -> fused scale/shift (deterministic,
// double accumulation, no atomics). stats[c]=a, stats[128+c]=b.
// ---------------------------------------------------------------------------
__global__ __launch_bounds__(256) void k2_stats(
    const float* __restrict__ tot, const float* __restrict__ gamma,
    const float* __restrict__ beta, float* __restrict__ stats)
{
    __shared__ double ssum[256];
    __shared__ double ssq[256];
    const int c   = blockIdx.x;
    const int tid = threadIdx.x;
    double s = 0.0, q = 0.0;
    for (int n = 0; n < N_; ++n) {
        const float* p = tot + (size_t)(n * COUT + c) * (T_ * V_);
        for (int i = tid; i < T_ * V_; i += 256) {
            const double v = (double)p[i];
            s += v;
            q += v * v;
        }
    }
    ssum[tid] = s; ssq[tid] = q;
    __syncthreads();
    for (int off = 128; off > 0; off >>= 1) {
        if (tid < off) {
            ssum[tid] += ssum[tid + off];
            ssq[tid]  += ssq[tid + off];
        }
        __syncthreads();
    }
    if (tid == 0) {
        const double cnt  = (double)N_ * T_ * V_;
        const double mean = ssum[0] / cnt;
        const double var  = ssq[0] / cnt - mean * mean;
        const float  a    = gamma[c] * rsqrtf((float)var + EPS_);
        const float  b    = beta[c] - (float)mean * a;
        stats[c]        = a;
        stats[COUT + c] = b;
    }
}

// ---------------------------------------------------------------------------
// Kernel 3: in-place normalize: y = y*a[c] + b[c]. One block per (n,c).
// ---------------------------------------------------------------------------
__global__ __launch_bounds__(256) void k3_norm(
    float* __restrict__ y, const float* __restrict__ stats)
{
    const int blk = blockIdx.x;       // n*128 + c
    const int c   = blk & (COUT - 1);
    const float a  = stats[c];
    const float bb = stats[COUT + c];
    float* p = y + (size_t)blk * (T_ * V_);
    for (int i = threadIdx.x; i < T_ * V_; i += 256)
        p[i] = p[i] * a + bb;
}

extern "C" void kernel_launch(void* const* d_in, const int* in_sizes, int n_in,
                              void* d_out, int out_size, void* d_ws, size_t ws_size,
                              hipStream_t stream)
{
    (void)in_sizes; (void)n_in; (void)out_size; (void)ws_size;
    const float* x     = (const float*)d_in[0];
    const float* A     = (const float*)d_in[1];
    const float* PA    = (const float*)d_in[2];
    const float* Wg    = (const float*)d_in[3];
    const float* bg    = (const float*)d_in[4];
    const float* gamma = (const float*)d_in[5];
    const float* beta  = (const float*)d_in[6];
    float* out   = (float*)d_out;
    float* stats = (float*)d_ws;   // 256 floats

    k1_fused<<<dim3(N_ * (T_ / TT)), dim3(256), 0, stream>>>(x, A, PA, Wg, bg, out);
    k2_stats<<<dim3(COUT), dim3(256), 0, stream>>>(out, gamma, beta, stats);
    k3_norm <<<dim3(N_ * COUT), dim3(256), 0, stream>>>(out, stats);
}